// EKF_283467841563
// MI455X (gfx1250) — compile-verified
//
#include <hip/hip_runtime.h>
#include <hip/hip_bf16.h>

// ---------------------------------------------------------------------------
// Batched discrete EKF propagate step on MI455X (gfx1250, wave32, WMMA).
// One wave32 per batch element; all 16x16 products via V_WMMA_F32_16X16X4_F32
// chained over K in 4-blocks. LDS tiles used for D-frag <-> A/B-frag layout
// conversion and transposes. 8x8 innovation inverse via lane-parallel
// Gauss-Jordan with __shfl pivot broadcast.
// ---------------------------------------------------------------------------

#define NXDIM 16
#define NZDIM 8
#define NUDIM 4

typedef __attribute__((ext_vector_type(2))) float v2f;
typedef __attribute__((ext_vector_type(8))) float v8f;

static __device__ __forceinline__ v8f vzero8() {
  v8f z = {0.f, 0.f, 0.f, 0.f, 0.f, 0.f, 0.f, 0.f};
  return z;
}

// D = A(16x4) x B(4x16) + C, f32, one wave.
static __device__ __forceinline__ v8f wmma4(v2f a, v2f b, v8f c) {
  // 8-arg pattern: (neg_a, A, neg_b, B, c_mod, C, reuse_a, reuse_b)
  return __builtin_amdgcn_wmma_f32_16x16x4_f32(
      false, a, false, b, (short)0, c, false, false);
}

// --- fragment loaders from a 16x16 row-major tile in LDS ---
// A-frag (16x4 block kb): lane l<16 holds rows M=l; halves supply K pairs.
static __device__ __forceinline__ v2f fragA(const float* T, int l, int h, int kb) {
  int k = 4 * kb + 2 * h;
  v2f r; r.x = T[l * 16 + k]; r.y = T[l * 16 + k + 1]; return r;
}
// A-frag of T^T
static __device__ __forceinline__ v2f fragAT(const float* T, int l, int h, int kb) {
  int k = 4 * kb + 2 * h;
  v2f r; r.x = T[k * 16 + l]; r.y = T[(k + 1) * 16 + l]; return r;
}
// B-frag (4x16 block kb): lane l<16 holds cols N=l; halves supply K pairs.
static __device__ __forceinline__ v2f fragB(const float* T, int l, int h, int kb) {
  int k = 4 * kb + 2 * h;
  v2f r; r.x = T[k * 16 + l]; r.y = T[(k + 1) * 16 + l]; return r;
}
// B-frag of T^T
static __device__ __forceinline__ v2f fragBT(const float* T, int l, int h, int kb) {
  int k = 4 * kb + 2 * h;
  v2f r; r.x = T[l * 16 + k]; r.y = T[l * 16 + k + 1]; return r;
}
// D/C layout: VGPR r = row (r + 8*h), col l.
static __device__ __forceinline__ void storeD(float* T, int l, int h, v8f d) {
#pragma unroll
  for (int r = 0; r < 8; ++r) T[(r + 8 * h) * 16 + l] = d[r];
}
static __device__ __forceinline__ v8f loadC(const float* T, int l, int h) {
  v8f c;
#pragma unroll
  for (int r = 0; r < 8; ++r) c[r] = T[(r + 8 * h) * 16 + l];
  return c;
}

// per-wave small-vector LDS slots
enum { RM = 0, RU = 16, RZ = 20, MB = 28, DM = 44, EV = 60, ZMH = 68, VLEN = 80 };

__global__ __launch_bounds__(256) void ekf_wmma_kernel(
    const float* __restrict__ g_mu, const float* __restrict__ g_Sig,
    const float* __restrict__ g_u, const float* __restrict__ g_z,
    const float* __restrict__ g_A, const float* __restrict__ g_Bm,
    const float* __restrict__ g_C, const float* __restrict__ g_Q,
    const float* __restrict__ g_R,
    float* __restrict__ out_mu, float* __restrict__ out_sig, int nbatch) {
  // block-shared system matrices
  __shared__ float Ash[256], Rsh[256], Csh[128], Qsh[64], Bmsh[64];
  // per-wave scratch tiles (layout conversion / transposes)
  __shared__ float t0s[8][256], t1s[8][256], t2s[8][256], t3s[8][256];
  __shared__ float wvs[8][VLEN];
  __shared__ float sis[8][64];

  const int tid = threadIdx.x;
  const int lane = tid & 31;
  const int w = tid >> 5;
  const int l = lane & 15;   // row (A/D) or col (B) index within tile
  const int h = lane >> 4;   // wave half selects K / row-group

  // cooperative load of shared system matrices
  Ash[tid] = g_A[tid];
  Rsh[tid] = g_R[tid];
  if (tid < 128) Csh[tid] = g_C[tid];
  if (tid < 64) { Qsh[tid] = g_Q[tid]; Bmsh[tid] = g_Bm[tid]; }
  __syncthreads();

  const int elem = blockIdx.x * 8 + w;
  if (elem >= nbatch) return;  // whole-wave uniform: EXEC stays all-ones

  float* T0 = t0s[w];
  float* T1 = t1s[w];
  float* T2 = t2s[w];
  float* T3 = t3s[w];
  float* V  = wvs[w];
  float* SI = sis[w];

  // ---- load Sigma_prev tile (coalesced b128) into T0, row-major ----
  {
    const float4* sp = (const float4*)(g_Sig + (size_t)elem * 256);
    float4 s0 = sp[lane];
    float4 s1 = sp[lane + 32];
    ((float4*)T0)[lane] = s0;
    ((float4*)T0)[lane + 32] = s1;
  }

  // ---- small-vector phase (intra-wave; LDS unit is in-order) ----
  if (lane < NXDIM) V[RM + lane] = g_mu[(size_t)elem * NXDIM + lane];
  if (lane < NZDIM) V[RZ + lane] = g_z[(size_t)elem * NZDIM + lane];
  if (lane < NUDIM) V[RU + lane] = g_u[(size_t)elem * NUDIM + lane];

  if (lane < NXDIM) {
    float pre = 0.f;
#pragma unroll
    for (int j = 0; j < NXDIM; ++j) pre += Ash[lane * 16 + j] * V[RM + j];
#pragma unroll
    for (int j = 0; j < NUDIM; ++j) pre += Bmsh[lane * 4 + j] * V[RU + j];
    float mb = tanhf(pre);
    V[MB + lane] = mb;               // mu_bar
    V[DM + lane] = 1.f - mb * mb;    // df/dpre diag -> G = diag(dm) * A
  }
  if (lane < NZDIM) {
    float hp = 0.f;
#pragma unroll
    for (int j = 0; j < NXDIM; ++j) hp += Csh[lane * 16 + j] * V[MB + j];
    float hb = tanhf(hp);
    V[EV + lane] = 1.f - hb * hb;    // H = diag(e) * C
    V[ZMH + lane] = V[RZ + lane] - hb;
  }

  const float dm_l = V[DM + l];                    // row/col scale for G
  const float e_l  = (l < NZDIM) ? V[EV + l] : 0.f;

  // ---- (1) GS = G * Sigma ----
  v8f acc = vzero8();
#pragma unroll
  for (int kb = 0; kb < 4; ++kb) {
    v2f a = fragA(Ash, l, h, kb) * dm_l;           // G[m,k] = dm[m]*A[m,k]
    v2f b = fragB(T0, l, h, kb);
    acc = wmma4(a, b, acc);
  }
  storeD(T1, l, h, acc);

  // ---- (2) Sig_bar = GS * G^T + R ----
  acc = loadC(Rsh, l, h);
#pragma unroll
  for (int kb = 0; kb < 4; ++kb) {
    v2f a = fragA(T1, l, h, kb);
    v2f b = fragBT(Ash, l, h, kb) * dm_l;          // G^T[k,n] = dm[n]*A[n,k]
    acc = wmma4(a, b, acc);
  }
  storeD(T2, l, h, acc);

  // ---- (3) tp = Sig_bar * H^T   (H^T padded 16x16, cols>=8 zero) ----
  acc = vzero8();
#pragma unroll
  for (int kb = 0; kb < 4; ++kb) {
    int k = 4 * kb + 2 * h;
    int cr = (l < NZDIM) ? l : 0;
    v2f a = fragA(T2, l, h, kb);
    v2f b;
    b.x = e_l * Csh[cr * 16 + k];                  // Ht[k,n] = e[n]*C[n,k]
    b.y = e_l * Csh[cr * 16 + k + 1];
    acc = wmma4(a, b, acc);
  }
  storeD(T1, l, h, acc);                           // tp

  // ---- (4) S = H * tp + Q  (padded 16x16, valid 8x8 top-left) ----
  v8f accS;
#pragma unroll
  for (int r = 0; r < 8; ++r) {
    int M = r + 8 * h;
    accS[r] = (M < NZDIM && l < NZDIM) ? Qsh[M * 8 + l] : 0.f;
  }
#pragma unroll
  for (int kb = 0; kb < 4; ++kb) {
    int k = 4 * kb + 2 * h;
    int cr = (l < NZDIM) ? l : 0;
    v2f a;
    a.x = e_l * Csh[cr * 16 + k];                  // H[m,k] = e[m]*C[m,k]
    a.y = e_l * Csh[cr * 16 + k + 1];
    v2f b = fragB(T1, l, h, kb);
    accS = wmma4(a, b, accS);
  }
  if (lane < 8) {                                  // rows 0..7 live in half 0
#pragma unroll
    for (int r = 0; r < 8; ++r) SI[r * 8 + lane] = accS[r];  // S[r][lane]
  }

  // ---- (5) Sinv via wave-parallel Gauss-Jordan (rows mirrored over halves) ----
  {
    const int rr = lane & 7;
    float row[16];
#pragma unroll
    for (int j = 0; j < 8; ++j) row[j] = SI[rr * 8 + j];
#pragma unroll
    for (int j = 0; j < 8; ++j) row[8 + j] = (j == rr) ? 1.f : 0.f;
#pragma unroll
    for (int kk = 0; kk < 8; ++kk) {
      float pk[16];
#pragma unroll
      for (int j = 0; j < 16; ++j) pk[j] = __shfl(row[j], kk, 32);
      float s = 1.f / pk[kk];
#pragma unroll
      for (int j = 0; j < 16; ++j) pk[j] *= s;
      if (rr == kk) {
#pragma unroll
        for (int j = 0; j < 16; ++j) row[j] = pk[j];
      } else {
        float f = row[kk];
#pragma unroll
        for (int j = 0; j < 16; ++j) row[j] -= f * pk[j];
      }
    }
    if (lane < 8) {
#pragma unroll
      for (int j = 0; j < 8; ++j) SI[rr * 8 + j] = row[8 + j];  // Sinv
    }
  }

  // ---- (6) K = tp * SinvPad ----
  acc = vzero8();
#pragma unroll
  for (int kb = 0; kb < 4; ++kb) {
    int k = 4 * kb + 2 * h;
    int kc0 = k & 7, kc1 = (k + 1) & 7, lc = (l < 8) ? l : 0;
    v2f a = fragA(T1, l, h, kb);
    v2f b;
    b.x = (k < 8 && l < 8) ? SI[kc0 * 8 + lc] : 0.f;
    b.y = (k + 1 < 8 && l < 8) ? SI[kc1 * 8 + lc] : 0.f;
    acc = wmma4(a, b, acc);
  }
  storeD(T3, l, h, acc);                           // K (cols>=8 zero)

  // ---- mu_now = mu_bar + K * (z - h) ----
  if (lane < NXDIM) {
    float m = V[MB + lane];
#pragma unroll
    for (int j = 0; j < NZDIM; ++j) m += T3[lane * 16 + j] * V[ZMH + j];
    out_mu[(size_t)elem * NXDIM + lane] = m;
  }

  // ---- (7) KH, then I_KH ----
  acc = vzero8();
#pragma unroll
  for (int kb = 0; kb < 4; ++kb) {
    int k = 4 * kb + 2 * h;
    int kc0 = k & 7, kc1 = (k + 1) & 7;
    v2f a = fragA(T3, l, h, kb);
    v2f b;
    b.x = (k < 8) ? V[EV + kc0] * Csh[kc0 * 16 + l] : 0.f;   // H[k,n]
    b.y = (k + 1 < 8) ? V[EV + kc1] * Csh[kc1 * 16 + l] : 0.f;
    acc = wmma4(a, b, acc);
  }
  v8f ikh;
#pragma unroll
  for (int r = 0; r < 8; ++r) {
    int M = r + 8 * h;
    ikh[r] = ((M == l) ? 1.f : 0.f) - acc[r];
  }
  storeD(T0, l, h, ikh);                           // I_KH overwrites Sigma tile

  // ---- (8) T1 = I_KH * Sig_bar ----
  acc = vzero8();
#pragma unroll
  for (int kb = 0; kb < 4; ++kb)
    acc = wmma4(fragA(T0, l, h, kb), fragB(T2, l, h, kb), acc);
  storeD(T1, l, h, acc);

  // ---- (9) sig_part = T1 * I_KH^T ----
  acc = vzero8();
#pragma unroll
  for (int kb = 0; kb < 4; ++kb)
    acc = wmma4(fragA(T1, l, h, kb), fragBT(T0, l, h, kb), acc);
  v8f sig_part = acc;

  // ---- (10) KQ = K * Qpad ----
  acc = vzero8();
#pragma unroll
  for (int kb = 0; kb < 4; ++kb) {
    int k = 4 * kb + 2 * h;
    int kc0 = k & 7, kc1 = (k + 1) & 7, lc = (l < 8) ? l : 0;
    v2f a = fragA(T3, l, h, kb);
    v2f b;
    b.x = (k < 8 && l < 8) ? Qsh[kc0 * 8 + lc] : 0.f;
    b.y = (k + 1 < 8 && l < 8) ? Qsh[kc1 * 8 + lc] : 0.f;
    acc = wmma4(a, b, acc);
  }
  storeD(T1, l, h, acc);                           // KQ

  // ---- (11) Sig_now = KQ * K^T + sig_part ----
  acc = sig_part;
#pragma unroll
  for (int kb = 0; kb < 4; ++kb)
    acc = wmma4(fragA(T1, l, h, kb), fragBT(T3, l, h, kb), acc);

  // ---- write Sigma_now directly from D-fragment ----
  float* os = out_sig + (size_t)elem * 256;
#pragma unroll
  for (int r = 0; r < 8; ++r) os[(r + 8 * h) * 16 + l] = acc[r];
}

extern "C" void kernel_launch(void* const* d_in, const int* in_sizes, int n_in,
                              void* d_out, int out_size, void* d_ws, size_t ws_size,
                              hipStream_t stream) {
  (void)n_in; (void)out_size; (void)d_ws; (void)ws_size;
  const float* mu_prev   = (const float*)d_in[0];
  const float* Sigma_prev= (const float*)d_in[1];
  const float* u         = (const float*)d_in[2];
  const float* z         = (const float*)d_in[3];
  const float* A         = (const float*)d_in[4];
  const float* Bm        = (const float*)d_in[5];
  const float* C         = (const float*)d_in[6];
  const float* Q         = (const float*)d_in[7];
  const float* R         = (const float*)d_in[8];

  const int nbatch = in_sizes[0] / NXDIM;          // B
  float* out_mu  = (float*)d_out;                  // [B,16]
  float* out_sig = out_mu + (size_t)nbatch * NXDIM;// [B,16,16]

  dim3 block(256);                                 // 8 wave32 per block
  dim3 grid((nbatch + 7) / 8);
  hipLaunchKernelGGL(ekf_wmma_kernel, grid, block, 0, stream,
                     mu_prev, Sigma_prev, u, z, A, Bm, C, Q, R,
                     out_mu, out_sig, nbatch);
}